// DGCNN_32512902431458
// MI455X (gfx1250) — compile-verified
//
#include <hip/hip_runtime.h>
#include <hip/hip_bf16.h>
#include <math.h>

typedef __attribute__((ext_vector_type(2))) float v2f;
typedef __attribute__((ext_vector_type(8))) float v8f;

#define NN  131072   // nodes
#define EE  4194304  // edges
#define GG  2048     // graphs
#define HH  32       // hidden
#define TLD 97       // total latent dim
#define KSP 30       // sort-pool k

// ---------------------------------------------------------------------------
// Generic f32 WMMA GEMM:  C[M,Nc] = A[M,K] @ B[K,Nc]   (row-major, one wave
// per 16x16 C tile, V_WMMA_F32_16X16X4_F32 K-loop). GUARD handles K%4!=0,
// BIASRELU fuses  C = relu(C + bias[n]).
// ---------------------------------------------------------------------------
template <bool GUARD, bool BIASRELU>
__global__ __launch_bounds__(256)
void wmma_gemm_f32(const float* __restrict__ A, const float* __restrict__ B,
                   const float* __restrict__ bias, float* __restrict__ C,
                   int M, int K, int Nc)
{
    const int wave = (blockIdx.x * blockDim.x + threadIdx.x) >> 5;
    const int lane = threadIdx.x & 31;
    const int ntn  = Nc >> 4;
    const int total = (M >> 4) * ntn;
    if (wave >= total) return;               // wave-uniform: EXEC stays all-1s

    const int tm   = (wave / ntn) << 4;
    const int tn   = (wave % ntn) << 4;
    const int half = lane >> 4;              // 0: K0/K1  1: K2/K3
    const int r    = lane & 15;              // row (A) / col (B) within tile

    v8f acc = {0.f, 0.f, 0.f, 0.f, 0.f, 0.f, 0.f, 0.f};
    const float* __restrict__ arow = A + (size_t)(tm + r) * K;

    for (int k0 = 0; k0 < K; k0 += 4) {
        const int ka = k0 + 2 * half;
        v2f a, b;
        if (GUARD) {
            a.x = (ka + 0 < K) ? arow[ka + 0] : 0.f;
            a.y = (ka + 1 < K) ? arow[ka + 1] : 0.f;
            b.x = (ka + 0 < K) ? B[(size_t)(ka + 0) * Nc + tn + r] : 0.f;
            b.y = (ka + 1 < K) ? B[(size_t)(ka + 1) * Nc + tn + r] : 0.f;
        } else {
            a.x = arow[ka + 0];
            a.y = arow[ka + 1];
            b.x = B[(size_t)(ka + 0) * Nc + tn + r];
            b.y = B[(size_t)(ka + 1) * Nc + tn + r];
        }
        acc = __builtin_amdgcn_wmma_f32_16x16x4_f32(
                  false, a, false, b, (short)0, acc, false, false);
    }

#pragma unroll
    for (int v = 0; v < 8; ++v) {            // C/D layout: M = v + 8*half
        const int m = tm + v + 8 * half;
        const int n = tn + r;
        float x = acc[v];
        if (BIASRELU) x = fmaxf(x + bias[n], 0.f);
        C[(size_t)m * Nc + n] = x;
    }
}

// ---------------------------------------------------------------------------
// Embedding:  h[n,c] = z_table[z[n], c]
// ---------------------------------------------------------------------------
__global__ void embed_k(const int* __restrict__ z, const float* __restrict__ tab,
                        float* __restrict__ h)
{
    int i = blockIdx.x * blockDim.x + threadIdx.x;
    if (i >= NN * HH) return;
    int node = i >> 5, c = i & 31;
    h[i] = tab[(size_t)z[node] * HH + c];
}

// ---------------------------------------------------------------------------
// Degree / norm:  deg = 1 + segsum(w, col);  dinv = deg^-1/2   (in-place)
// ---------------------------------------------------------------------------
__global__ void deg_init_k(float* __restrict__ deg)
{
    int i = blockIdx.x * blockDim.x + threadIdx.x;
    if (i < NN) deg[i] = 1.0f;               // self-loop weight
}
__global__ void deg_accum_k(const int* __restrict__ ei, const float* __restrict__ w,
                            float* __restrict__ deg)
{
    int e = blockIdx.x * blockDim.x + threadIdx.x;
    if (e >= EE) return;
    unsafeAtomicAdd(&deg[ei[EE + e]], w[e]);
}
__global__ void dinv_fin_k(float* __restrict__ deg)
{
    int i = blockIdx.x * blockDim.x + threadIdx.x;
    if (i < NN) deg[i] = rsqrtf(deg[i]);     // deg >= 1 always
}

// ---------------------------------------------------------------------------
// Self-loop message:  agg[n,c] = dinv[n]^2 * xw[n,c]
// ---------------------------------------------------------------------------
__global__ void self_init_k(const float* __restrict__ xw, const float* __restrict__ dinv,
                            float* __restrict__ agg)
{
    int i = blockIdx.x * blockDim.x + threadIdx.x;
    if (i >= NN * HH) return;
    float d = dinv[i >> 5];
    agg[i] = d * d * xw[i];
}

// ---------------------------------------------------------------------------
// Edge scatter (32 channels): one wave32 per edge, lane = channel.
// Coalesced 128B gather of xw[row,:], 32 native f32 atomics into agg[col,:].
// ---------------------------------------------------------------------------
__global__ __launch_bounds__(256)
void edge_scatter32_k(const int* __restrict__ ei, const float* __restrict__ w,
                      const float* __restrict__ dinv, const float* __restrict__ xw,
                      float* __restrict__ agg)
{
    int t = blockIdx.x * blockDim.x + threadIdx.x;
    int e = t >> 5;
    int c = threadIdx.x & 31;
    if (e >= EE) return;
    if (c == 0) __builtin_prefetch(ei + e + 4096, 0, 1);   // stream edge idx
    int rs = ei[e];
    int cd = ei[EE + e];
    float nrm = dinv[rs] * w[e] * dinv[cd];
    unsafeAtomicAdd(&agg[(size_t)cd * HH + c], nrm * xw[(size_t)rs * HH + c]);
}

// ---------------------------------------------------------------------------
// Activation + concat write:  h[n,c] = tanh(agg[n,c] + b[c]); xcat[n,off+c]=h
// ---------------------------------------------------------------------------
__global__ void act_store_k(const float* __restrict__ agg, const float* __restrict__ b,
                            float* __restrict__ h, float* __restrict__ xcat, int off)
{
    int i = blockIdx.x * blockDim.x + threadIdx.x;
    if (i >= NN * HH) return;
    int node = i >> 5, c = i & 31;
    float v = tanhf(agg[i] + b[c]);
    h[i] = v;
    xcat[(size_t)node * TLD + off + c] = v;
}

// ---------------------------------------------------------------------------
// Layer 3 (H -> 1)
// ---------------------------------------------------------------------------
__global__ void xw3_k(const float* __restrict__ h, const float* __restrict__ W3,
                      float* __restrict__ xw3)
{
    int n = blockIdx.x * blockDim.x + threadIdx.x;
    if (n >= NN) return;
    float s = 0.f;
#pragma unroll
    for (int k = 0; k < HH; ++k) s += h[(size_t)n * HH + k] * W3[k];
    xw3[n] = s;
}
__global__ void self_init1_k(const float* __restrict__ xw3, const float* __restrict__ dinv,
                             float* __restrict__ agg3)
{
    int n = blockIdx.x * blockDim.x + threadIdx.x;
    if (n >= NN) return;
    float d = dinv[n];
    agg3[n] = d * d * xw3[n];
}
__global__ void edge_scatter1_k(const int* __restrict__ ei, const float* __restrict__ w,
                                const float* __restrict__ dinv, const float* __restrict__ xw3,
                                float* __restrict__ agg3)
{
    int e = blockIdx.x * blockDim.x + threadIdx.x;
    if (e >= EE) return;
    int rs = ei[e];
    int cd = ei[EE + e];
    unsafeAtomicAdd(&agg3[cd], dinv[rs] * w[e] * dinv[cd] * xw3[rs]);
}
__global__ void act3_k(const float* __restrict__ agg3, const float* __restrict__ b3,
                       float* __restrict__ xcat)
{
    int n = blockIdx.x * blockDim.x + threadIdx.x;
    if (n >= NN) return;
    xcat[(size_t)n * TLD + 96] = tanhf(agg3[n] + b3[0]);
}

// ---------------------------------------------------------------------------
// Sort-pool: per graph (64 nodes) stable descending rank by last channel;
// unique ranks -> conflict-free scatter of top-30 full rows [30,97].
// ---------------------------------------------------------------------------
__global__ __launch_bounds__(64)
void sortpool_k(const float* __restrict__ xcat, float* __restrict__ pooled)
{
    __shared__ float vals[64];
    int g = blockIdx.x, t = threadIdx.x;
    int node = g * 64 + t;
    float v = xcat[(size_t)node * TLD + 96];
    vals[t] = v;
    __syncthreads();
    int rank = 0;
    for (int j = 0; j < 64; ++j) {
        float vj = vals[j];
        rank += (vj > v) || (vj == v && j < t);   // stable (index asc) tiebreak
    }
    if (rank < KSP) {
        const float* __restrict__ src = xcat + (size_t)node * TLD;
        float* __restrict__ dst = pooled + ((size_t)g * KSP + rank) * TLD;
        for (int c = 0; c < TLD; ++c) dst[c] = src[c];
    }
}

// ---------------------------------------------------------------------------
// conv1 weight transpose: Bt[t,c] = conv1_w[c,0,t]   (97x16)
// ---------------------------------------------------------------------------
__global__ void conv1_tr_k(const float* __restrict__ w1, float* __restrict__ bt)
{
    int i = blockIdx.x * blockDim.x + threadIdx.x;
    if (i >= TLD * 16) return;
    int t = i >> 4, c = i & 15;
    bt[i] = w1[(size_t)c * TLD + t];
}

// ---------------------------------------------------------------------------
// Max-pool (window 2 stride 2 over k): y1 [g*30+k, 16] -> y1p [g*15+j, 16]
// ---------------------------------------------------------------------------
__global__ void maxpool_k(const float* __restrict__ y1, float* __restrict__ y1p)
{
    int i = blockIdx.x * blockDim.x + threadIdx.x;
    if (i >= GG * 15 * 16) return;
    int c = i & 15;
    int j = (i >> 4) % 15;
    int g = i / (15 * 16);
    const float* base = y1 + ((size_t)g * KSP + 2 * j) * 16 + c;
    y1p[i] = fmaxf(base[0], base[16]);
}

// ---------------------------------------------------------------------------
// conv2 (16ch x k5, stride 1) + relu; flatten in reference order o*11+j
// ---------------------------------------------------------------------------
__global__ void conv2_k(const float* __restrict__ y1p, const float* __restrict__ w2,
                        const float* __restrict__ b2, float* __restrict__ y2)
{
    int i = blockIdx.x * blockDim.x + threadIdx.x;
    if (i >= GG * 32 * 11) return;
    int j = i % 11;
    int o = (i / 11) & 31;
    int g = i / (11 * 32);
    float s = b2[o];
    for (int t = 0; t < 5; ++t) {
        const float* __restrict__ row = y1p + ((size_t)g * 15 + j + t) * 16;
        const float* __restrict__ wr  = w2 + (size_t)o * 16 * 5 + t;  // w2[o][ci][t]
#pragma unroll
        for (int ci = 0; ci < 16; ++ci) s += row[ci] * wr[ci * 5];
    }
    y2[(size_t)g * 352 + o * 11 + j] = fmaxf(s, 0.f);
}

// ---------------------------------------------------------------------------
// lin2: out[g] = fc1[g,:] @ lin2_w + b
// ---------------------------------------------------------------------------
__global__ void lin2_k(const float* __restrict__ fc1, const float* __restrict__ w,
                       const float* __restrict__ b, float* __restrict__ out)
{
    int g = blockIdx.x * blockDim.x + threadIdx.x;
    if (g >= GG) return;
    float s = b[0];
#pragma unroll 8
    for (int k = 0; k < 128; ++k) s += fc1[(size_t)g * 128 + k] * w[k];
    out[g] = s;
}

// ---------------------------------------------------------------------------
extern "C" void kernel_launch(void* const* d_in, const int* in_sizes, int n_in,
                              void* d_out, int out_size, void* d_ws, size_t ws_size,
                              hipStream_t stream)
{
    const int*   z      = (const int*)  d_in[0];
    const int*   ei     = (const int*)  d_in[1];   // [2,E] row-major
    const float* ew     = (const float*)d_in[2];
    const float* ztab   = (const float*)d_in[4];
    const float* Wl[4]  = {(const float*)d_in[5], (const float*)d_in[7],
                           (const float*)d_in[9], (const float*)d_in[11]};
    const float* bl[4]  = {(const float*)d_in[6], (const float*)d_in[8],
                           (const float*)d_in[10], (const float*)d_in[12]};
    const float* c1w    = (const float*)d_in[13];
    const float* c1b    = (const float*)d_in[14];
    const float* c2w    = (const float*)d_in[15];
    const float* c2b    = (const float*)d_in[16];
    const float* l1w    = (const float*)d_in[17];
    const float* l1b    = (const float*)d_in[18];
    const float* l2w    = (const float*)d_in[19];
    const float* l2b    = (const float*)d_in[20];
    float* out = (float*)d_out;

    // workspace layout (floats)
    float* w      = (float*)d_ws;
    float* f_h    = w;                                   // N*32
    float* f_xw   = f_h    + (size_t)NN * HH;            // N*32
    float* f_agg  = f_xw   + (size_t)NN * HH;            // N*32
    float* f_dinv = f_agg  + (size_t)NN * HH;            // N (deg -> dinv)
    float* f_xw3  = f_dinv + NN;                         // N
    float* f_agg3 = f_xw3  + NN;                         // N
    float* f_xcat = f_agg3 + NN;                         // N*97
    float* f_pool = f_xcat + (size_t)NN * TLD;           // G*30*97
    float* f_y1   = f_pool + (size_t)GG * KSP * TLD;     // G*30*16
    float* f_y1p  = f_y1   + (size_t)GG * KSP * 16;      // G*15*16
    float* f_y2   = f_y1p  + (size_t)GG * 15 * 16;       // G*352
    float* f_fc1  = f_y2   + (size_t)GG * 352;           // G*128
    float* f_bt   = f_fc1  + (size_t)GG * 128;           // 97*16

    const int B = 256;
    const int gNH = (NN * HH + B - 1) / B;       // N*32 elementwise
    const int gN  = (NN + B - 1) / B;
    const int gE  = (EE + B - 1) / B;
    const int gE32 = EE * 32 / B;                // wave-per-edge

    // 1. embedding
    embed_k<<<gNH, B, 0, stream>>>(z, ztab, f_h);

    // 2. degree -> dinv
    deg_init_k<<<gN, B, 0, stream>>>(f_dinv);
    deg_accum_k<<<gE, B, 0, stream>>>(ei, ew, f_dinv);
    dinv_fin_k<<<gN, B, 0, stream>>>(f_dinv);

    // 3. GCN layers 0..2 (H -> H)
    for (int l = 0; l < 3; ++l) {
        int tiles  = (NN / 16) * (HH / 16);       // 16384
        int blocks = (tiles * 32 + B - 1) / B;
        wmma_gemm_f32<false, false><<<blocks, B, 0, stream>>>(
            f_h, Wl[l], nullptr, f_xw, NN, HH, HH);
        self_init_k<<<gNH, B, 0, stream>>>(f_xw, f_dinv, f_agg);
        edge_scatter32_k<<<gE32, B, 0, stream>>>(ei, ew, f_dinv, f_xw, f_agg);
        act_store_k<<<gNH, B, 0, stream>>>(f_agg, bl[l], f_h, f_xcat, l * HH);
    }

    // 4. GCN layer 3 (H -> 1)
    xw3_k<<<gN, B, 0, stream>>>(f_h, Wl[3], f_xw3);
    self_init1_k<<<gN, B, 0, stream>>>(f_xw3, f_dinv, f_agg3);
    edge_scatter1_k<<<gE, B, 0, stream>>>(ei, ew, f_dinv, f_xw3, f_agg3);
    act3_k<<<gN, B, 0, stream>>>(f_agg3, bl[3], f_xcat);

    // 5. global sort pool (top-30 of 64 by last channel, stable desc)
    sortpool_k<<<GG, 64, 0, stream>>>(f_xcat, f_pool);

    // 6. conv1 (stride TL == per-node linear) as WMMA GEMM + bias + relu
    conv1_tr_k<<<(TLD * 16 + B - 1) / B, B, 0, stream>>>(c1w, f_bt);
    {
        int M = GG * KSP;                         // 61440
        int tiles  = (M / 16) * (16 / 16);        // 3840
        int blocks = (tiles * 32 + B - 1) / B;
        wmma_gemm_f32<true, true><<<blocks, B, 0, stream>>>(
            f_pool, f_bt, c1b, f_y1, M, TLD, 16);
    }

    // 7. maxpool k=2 s=2
    maxpool_k<<<(GG * 15 * 16 + B - 1) / B, B, 0, stream>>>(f_y1, f_y1p);

    // 8. conv2 + relu (+ flatten in reference order)
    conv2_k<<<(GG * 32 * 11 + B - 1) / B, B, 0, stream>>>(f_y1p, c2w, c2b, f_y2);

    // 9. lin1 (352 -> 128) as WMMA GEMM + bias + relu
    {
        int tiles  = (GG / 16) * (128 / 16);      // 1024
        int blocks = (tiles * 32 + B - 1) / B;
        wmma_gemm_f32<false, true><<<blocks, B, 0, stream>>>(
            f_y2, l1w, l1b, f_fc1, GG, 352, 128);
    }

    // 10. lin2 (128 -> 1)
    lin2_k<<<(GG + B - 1) / B, B, 0, stream>>>(f_fc1, l2w, l2b, out);
}